// SpatioConvLayer_66443144069634
// MI455X (gfx1250) — compile-verified
//
#include <hip/hip_runtime.h>
#include <hip/hip_bf16.h>

// Problem constants (from the reference)
#define B_ 8
#define C_ 32
#define T_ 12
#define N_ 5000
#define E_ 80000
#define F_ (B_ * C_ * T_)      // 3072 contiguous feature slices of length N_
#define SL 4                   // feature slices per workgroup (4*2*20KB = 160KB LDS)
#define NT ((N_ + 15) / 16)    // 313 node tiles of 16
#define TOT (B_ * T_ * NT)     // 30048 WMMA wave-tiles

typedef __attribute__((ext_vector_type(2))) float v2f;
typedef __attribute__((ext_vector_type(8))) float v8f;

// ---------------- kernel 1: zero degree buffers ----------------
__global__ void k_zero(float* __restrict__ ws) {
  int i = blockIdx.x * blockDim.x + threadIdx.x;
  if (i < 2 * N_) ws[i] = 0.0f;
}

// ---------------- kernel 2: accumulate degrees (global f32 atomics) ----------------
__global__ void k_deg(const int* __restrict__ src, const int* __restrict__ dst,
                      float* __restrict__ ws) {
  int e = blockIdx.x * blockDim.x + threadIdx.x;
  if (e < E_) {
    atomicAdd(&ws[src[e]], 1.0f);       // deg_out
    atomicAdd(&ws[N_ + dst[e]], 1.0f);  // deg_in
  }
}

// ---------------- kernel 3: degrees -> norms in place ----------------
__global__ void k_norm(float* __restrict__ ws) {
  int i = blockIdx.x * blockDim.x + threadIdx.x;
  if (i < 2 * N_) {
    ws[i] = rsqrtf(fmaxf(ws[i], 1.0f));
  }
}

// ---------------- kernel 4: LDS-resident edge scatter-add ----------------
// For SL feature slices f (each a contiguous N_-float array of x):
//   agg[f][n] = norm_dst[n] * sum_{e: dst[e]==n} norm_src[src[e]] * x[f][src[e]]
// Both gather source and scatter destination live in LDS (ds_add_f32 atomics).
// Result written to `agg` (aliases d_out; consumed by the WMMA kernel).
__global__ void __launch_bounds__(256)
k_agg(const float* __restrict__ x, const int* __restrict__ src,
      const int* __restrict__ dst, const float* __restrict__ ws,
      float* __restrict__ agg) {
  extern __shared__ float smem[];        // SL*N_ (xs) + SL*N_ (as)
  float* xs = smem;
  float* as = smem + SL * N_;
  const float* norm_src = ws;
  const float* norm_dst = ws + N_;
  const int f0 = blockIdx.x * SL;
  const int tid = threadIdx.x;

  // Stage 1: load x slices scaled by norm_src; zero accumulators.
  for (int n = tid; n < N_; n += 256) {
    float ns = norm_src[n];
#pragma unroll
    for (int j = 0; j < SL; ++j) {
      xs[j * N_ + n] = x[(long)(f0 + j) * N_ + n] * ns;
      as[j * N_ + n] = 0.0f;
    }
  }
  __syncthreads();

  // Stage 2: edge loop. One index-pair load feeds SL LDS float atomics.
  for (int e = tid; e < E_; e += 256) {
    __builtin_prefetch(&src[e + 2048], 0, 1);   // global_prefetch_b8
    int s = src[e];
    int d = dst[e];
#pragma unroll
    for (int j = 0; j < SL; ++j) {
      atomicAdd(&as[j * N_ + d], xs[j * N_ + s]);  // ds_add_f32
    }
  }
  __syncthreads();

  // Stage 3: scale by norm_dst and write out.
  for (int n = tid; n < N_; n += 256) {
    float nd = norm_dst[n];
#pragma unroll
    for (int j = 0; j < SL; ++j) {
      agg[(long)(f0 + j) * N_ + n] = as[j * N_ + n] * nd;
    }
  }
}

// ---------------- kernel 5: per-node 32x32 channel mix via WMMA f32 16x16x4 ----------------
// out[b][d][t][n] = relu( sum_c agg[b][c][t][n]*W[c][d] + bias[d] + x[b][d][t][n] )
// One wave owns a (b,t,16-node) tile: 8 K-steps x 2 d-blocks = 16 v_wmma_f32_16x16x4_f32.
// `agg` aliases `out`; each wave reads all its c-channel inputs before writing its
// d-channel outputs at the same addresses (in-wave data dependency => safe).
__global__ void __launch_bounds__(256)
k_gemm(const float* __restrict__ agg, const float* __restrict__ Wm,
       const float* __restrict__ bias, const float* __restrict__ x,
       float* __restrict__ out) {
  int wave = (blockIdx.x * 256 + (int)threadIdx.x) >> 5;  // wave-uniform
  if (wave >= TOT) return;                                 // whole-wave exit, EXEC stays full
  const int lane = threadIdx.x & 31;
  const int half = lane >> 4;   // 0: lanes 0-15, 1: lanes 16-31
  const int l16  = lane & 15;

  const int nblk = wave % NT;
  const int bt   = wave / NT;
  const int t = bt % T_;
  const int b = bt / T_;
  const int n0 = nblk * 16;

  // Preload W tiles in B-matrix layout (4x16 f32, K split across wave halves).
  v2f wreg[2][8];
#pragma unroll
  for (int db = 0; db < 2; ++db) {
#pragma unroll
    for (int kk = 0; kk < 8; ++kk) {
      int c0 = kk * 4 + 2 * half;
      int d  = db * 16 + l16;
      wreg[db][kk].x = Wm[(c0 + 0) * C_ + d];
      wreg[db][kk].y = Wm[(c0 + 1) * C_ + d];
    }
  }

  // A-matrix: M = l16 (node within tile), K pairs split across halves.
  int nA = n0 + l16;
  if (nA > N_ - 1) nA = N_ - 1;   // clamp tail tile; bogus rows never stored

  v8f acc0 = {};
  v8f acc1 = {};
#pragma unroll
  for (int kk = 0; kk < 8; ++kk) {
    int c0 = kk * 4 + 2 * half;
    v2f a;
    a.x = agg[((long)(b * C_ + c0 + 0) * T_ + t) * N_ + nA];
    a.y = agg[((long)(b * C_ + c0 + 1) * T_ + t) * N_ + nA];
    acc0 = __builtin_amdgcn_wmma_f32_16x16x4_f32(
        false, a, false, wreg[0][kk], (short)0, acc0, false, false);
    acc1 = __builtin_amdgcn_wmma_f32_16x16x4_f32(
        false, a, false, wreg[1][kk], (short)0, acc1, false, false);
  }

  // Epilogue: bias + residual + relu. D layout: VGPR v -> M = v + 8*half, N = l16.
  float bias0 = bias[l16];
  float bias1 = bias[16 + l16];
#pragma unroll
  for (int v = 0; v < 8; ++v) {
    int m = v + 8 * half;
    int n = n0 + m;
    if (n < N_) {
      long i0 = ((long)(b * C_ + l16) * T_ + t) * N_ + n;
      long i1 = ((long)(b * C_ + 16 + l16) * T_ + t) * N_ + n;
      float v0 = acc0[v] + bias0 + x[i0];
      float v1 = acc1[v] + bias1 + x[i1];
      out[i0] = fmaxf(v0, 0.0f);
      out[i1] = fmaxf(v1, 0.0f);
    }
  }
}

extern "C" void kernel_launch(void* const* d_in, const int* in_sizes, int n_in,
                              void* d_out, int out_size, void* d_ws, size_t ws_size,
                              hipStream_t stream) {
  (void)in_sizes; (void)n_in; (void)out_size; (void)ws_size;
  const float* x    = (const float*)d_in[0];  // [B,C,T,N]
  const float* Wm   = (const float*)d_in[1];  // [C,C]
  const float* bias = (const float*)d_in[2];  // [C]
  const int*   src  = (const int*)d_in[3];    // [E]
  const int*   dst  = (const int*)d_in[4];    // [E]
  float* out = (float*)d_out;                 // [B,C,T,N]
  float* ws  = (float*)d_ws;                  // 2*N_ floats: norm_src, norm_dst

  k_zero<<<(2 * N_ + 255) / 256, 256, 0, stream>>>(ws);
  k_deg<<<(E_ + 255) / 256, 256, 0, stream>>>(src, dst, ws);
  k_norm<<<(2 * N_ + 255) / 256, 256, 0, stream>>>(ws);

  size_t lds_bytes = (size_t)SL * 2 * N_ * sizeof(float);  // 160 KB (< 320 KB/WGP)
  k_agg<<<F_ / SL, 256, lds_bytes, stream>>>(x, src, dst, ws, out);

  int blocks = (TOT + 7) / 8;  // 8 waves of 32 per 256-thread block
  k_gemm<<<blocks, 256, 0, stream>>>(out, Wm, bias, x, out);
}